// ChipletThermalModel_23149873725604
// MI455X (gfx1250) — compile-verified
//
#include <hip/hip_runtime.h>
#include <stdint.h>

// ---------------------------------------------------------------------------
// Chiplet thermal model, MI455X (gfx1250, wave32).
// Compute-bound on transcendentals; WMMA f32 16x16x4 performs the P-weighted
// chiplet reduction; async-to-LDS staging of chiplet params when available.
// ---------------------------------------------------------------------------

typedef __attribute__((ext_vector_type(2))) float v2f;
typedef __attribute__((ext_vector_type(8))) float v8f;

#define AS1 __attribute__((address_space(1)))
#define AS3 __attribute__((address_space(3)))

#if defined(__has_builtin)
#  if __has_builtin(__builtin_amdgcn_global_load_async_to_lds_b32)
#    define USE_ASYNC_LDS 1
#  endif
#endif
#ifndef USE_ASYNC_LDS
#  define USE_ASYNC_LDS 0
#endif

#define LN2f   0.69314718055994531f
#define HLN2f  0.34657359027997265f   // 0.5*ln2
#define PI2f   1.57079632679489662f
#define TOSPIf 1.12837916709551257f   // 2/sqrt(pi)

// One corner of F(a,b,c); lb = log2(a2+b^2), lc = log2(a2+c^2) precomputed
// (shared across corners). Branchless atan with a single v_rcp.
__device__ __forceinline__ float corner_term(float b, float c, float lb, float lc,
                                             float a_s, float a2) {
  float d2    = __builtin_fmaf(b, b, __builtin_fmaf(c, c, a2));
  float delta = __builtin_amdgcn_sqrtf(d2);
  float t1 = b * (LN2f * __builtin_amdgcn_logf(c + delta) - HLN2f * lb);
  float t2 = c * (LN2f * __builtin_amdgcn_logf(b + delta) - HLN2f * lc);
  // t3 = a * atan(b*c / (a*delta)); a>0, delta>0
  float num = b * c;
  float den = a_s * delta;
  float an  = __builtin_fabsf(num);
  float mn  = __builtin_fminf(an, den);
  float mx  = __builtin_fmaxf(an, den);
  float t   = mn * __builtin_amdgcn_rcpf(mx);   // in [0,1]
  float s   = t * t;
  float p   = __builtin_fmaf(s, 0.0208351f, -0.0851330f);
  p = __builtin_fmaf(s, p, 0.1801410f);
  p = __builtin_fmaf(s, p, -0.3302995f);
  p = __builtin_fmaf(s, p, 0.9998660f);
  float at  = t * p;                            // atan(t), t<=1
  float atu = (an <= den) ? at : (PI2f - at);   // atan(|u|)
  float t3  = a_s * __builtin_copysignf(atu, num);
  return t1 + t2 - t3;
}

// Full contribution of chiplet `c` at point (px,py): P_c * (Boff + K*sum(F)).
__device__ __forceinline__ float eval_chiplet(float px, float py,
                                              const float* sh, int c,
                                              float a_s, float a2, float Boff) {
  float cx  = sh[0 * 16 + c];
  float cy  = sh[1 * 16 + c];
  float w2  = 0.5f * sh[2 * 16 + c];
  float h2  = 0.5f * sh[3 * 16 + c];
  float P   = sh[4 * 16 + c];
  float ilx = __builtin_amdgcn_rcpf(sh[5 * 16 + c]);
  float ily = __builtin_amdgcn_rcpf(sh[6 * 16 + c]);
  float dx = px - cx, dy = py - cy;
  float b0 = (w2 - dx) * ilx;   // sx = +1
  float b1 = (w2 + dx) * ilx;   // sx = -1
  float c0 = (h2 - dy) * ily;   // sy = +1
  float c1 = (h2 + dy) * ily;   // sy = -1
  // logs shared across the 4 corners: only 2 distinct b's and 2 distinct c's
  float lb0 = __builtin_amdgcn_logf(__builtin_fmaf(b0, b0, a2));
  float lb1 = __builtin_amdgcn_logf(__builtin_fmaf(b1, b1, a2));
  float lc0 = __builtin_amdgcn_logf(__builtin_fmaf(c0, c0, a2));
  float lc1 = __builtin_amdgcn_logf(__builtin_fmaf(c1, c1, a2));
  float raw = corner_term(b0, c0, lb0, lc0, a_s, a2);
  raw      += corner_term(b0, c1, lb0, lc1, a_s, a2);
  raw      += corner_term(b1, c0, lb1, lc0, a_s, a2);
  raw      += corner_term(b1, c1, lb1, lc1, a_s, a2);
  return P * __builtin_fmaf(TOSPIf, raw, Boff);
}

__global__ __launch_bounds__(256)
void ChipletThermal_wmma_kernel(const float* __restrict__ gx, const float* __restrict__ gy,
                                const float* __restrict__ ccx, const float* __restrict__ ccy,
                                const float* __restrict__ cww, const float* __restrict__ chh,
                                const float* __restrict__ cpp, const float* __restrict__ Aptr,
                                const float* __restrict__ aptr, const float* __restrict__ bptr,
                                const float* __restrict__ glx, const float* __restrict__ gly,
                                float* __restrict__ out, int G2, int tiles) {
  __shared__ float sh[7 * 16];  // [param][chiplet]: cx, cy, w, h, P, lx, ly
  const int t   = threadIdx.x;
  const int row = blockIdx.y;

  // ---- stage chiplet params to LDS (async on CDNA5 when available) ----
  if (t < 112) {
    int pp = t >> 4, c = t & 15;
    const float* s;
    switch (pp) {
      case 0:  s = ccx + row * 16 + c; break;
      case 1:  s = ccy + row * 16 + c; break;
      case 2:  s = cww + row * 16 + c; break;
      case 3:  s = chh + row * 16 + c; break;
      case 4:  s = cpp + row * 16 + c; break;
      case 5:  s = glx + c;            break;
      default: s = gly + c;            break;
    }
#if USE_ASYNC_LDS
    __builtin_amdgcn_global_load_async_to_lds_b32(
        (AS1 int*)(uint64_t)(uintptr_t)s,
        (AS3 int*)(uint32_t)(uintptr_t)&sh[t], 0, 0);
#else
    sh[t] = *s;
#endif
  }
#if USE_ASYNC_LDS
#  if defined(__has_builtin) && __has_builtin(__builtin_amdgcn_s_wait_asynccnt)
  __builtin_amdgcn_s_wait_asynccnt(0);
#  else
  asm volatile("s_wait_asynccnt 0" ::: "memory");
#  endif
#endif
  __syncthreads();

  const int lane = t & 31;
  const int tile = blockIdx.x * 8 + (t >> 5);  // one 16-point tile per wave
  if (tile >= tiles) return;                   // wave-uniform: EXEC stays all-1s

  const int p    = lane & 15;                  // point row M within the tile
  const int half = lane >> 4;                  // which K-pair this lane feeds
  const float A_s  = Aptr[0];
  const float a_s  = aptr[0];
  const float Boff = bptr[0];
  const float a2   = a_s * a_s;

  const int idx  = row * G2 + tile * 16 + p;
  const float px = gx[idx];
  const float py = gy[idx];

  // D = sum_k A[p][k] * 1.0 accumulated over 4 WMMAs (K=4 each, 16 chiplets).
  // A-matrix layout (ISA 7.12.2): lanes 0-15 -> K=0,1 ; lanes 16-31 -> K=2,3.
  // B == all-ones makes the result independent of B's exact K layout.
  v8f acc = {0.f, 0.f, 0.f, 0.f, 0.f, 0.f, 0.f, 0.f};
  v2f ones; ones.x = 1.0f; ones.y = 1.0f;
#pragma unroll
  for (int r = 0; r < 4; ++r) {
    const int c0 = r * 4 + half * 2;
    v2f av;
    av.x = eval_chiplet(px, py, sh, c0,     a_s, a2, Boff);
    av.y = eval_chiplet(px, py, sh, c0 + 1, a_s, a2, Boff);
    acc = __builtin_amdgcn_wmma_f32_16x16x4_f32(
        /*neg_a=*/false, av, /*neg_b=*/false, ones,
        /*c_mod=*/(short)0, acc, /*reuse_a=*/false, /*reuse_b=*/false);
  }

  // C/D layout: VGPR r holds M=r (lanes 0-15) / M=8+r (lanes 16-31), any N.
  // Point m<8  lives in acc[m]   of lanes 0-15;
  // point m>=8 lives in acc[m-8] of lanes 16-31.
  const int k7 = lane & 7;
  float v = acc[0];
  v = (k7 == 1) ? acc[1] : v;
  v = (k7 == 2) ? acc[2] : v;
  v = (k7 == 3) ? acc[3] : v;
  v = (k7 == 4) ? acc[4] : v;
  v = (k7 == 5) ? acc[5] : v;
  v = (k7 == 6) ? acc[6] : v;
  v = (k7 == 7) ? acc[7] : v;

  const bool writer = (lane < 8) || (lane >= 16 && lane < 24);
  if (writer) {
    const int pout = (lane < 16) ? lane : (lane - 8);
    out[row * G2 + tile * 16 + pout] = A_s * v;
  }
}

extern "C" void kernel_launch(void* const* d_in, const int* in_sizes, int n_in,
                              void* d_out, int out_size, void* d_ws, size_t ws_size,
                              hipStream_t stream) {
  (void)n_in; (void)out_size; (void)d_ws; (void)ws_size;
  const float* x    = (const float*)d_in[0];
  const float* y    = (const float*)d_in[1];
  const float* ccx  = (const float*)d_in[2];
  const float* ccy  = (const float*)d_in[3];
  const float* cw   = (const float*)d_in[4];
  const float* ch   = (const float*)d_in[5];
  const float* cp   = (const float*)d_in[6];
  const float* A    = (const float*)d_in[7];
  const float* a    = (const float*)d_in[8];
  const float* Boff = (const float*)d_in[9];
  const float* lx   = (const float*)d_in[10];
  const float* ly   = (const float*)d_in[11];
  float* out        = (float*)d_out;

  const int N     = in_sizes[10];          // 16 (kernel mapping assumes 16)
  const int Brows = in_sizes[2] / N;       // 64
  const int G2    = in_sizes[0] / Brows;   // 65536
  const int tiles = G2 / 16;               // 16-point tiles per batch row
  const int bx    = (tiles + 7) / 8;       // 8 waves (tiles) per block

  dim3 grid(bx, Brows), block(256);
  ChipletThermal_wmma_kernel<<<grid, block, 0, stream>>>(
      x, y, ccx, ccy, cw, ch, cp, A, a, Boff, lx, ly, out, G2, tiles);
}